// AHA_25074019074660
// MI455X (gfx1250) — compile-verified
//
#include <hip/hip_runtime.h>
#include <hip/hip_bf16.h>

// ---------------------------------------------------------------------------
// MI455X notes:
//  - Memory-bound problem: x is streamed twice (~680 MB total) -> ~29us at
//    23.3 TB/s. Kernels 1 & 4 are the streaming kernels.
//  - conv_down is a [64x256]x[256x150] fp32 GEMM per sample -> mapped onto
//    V_WMMA_F32_16X16X4_F32 (wave32, one wave per 16x16 output tile).
//  - B-tile bounds handling uses address clamping instead of exec-masked
//    loads: garbage only reaches D columns that are never stored, so the
//    hot loop has zero divergence (no s_and_saveexec per load).
// ---------------------------------------------------------------------------

typedef __attribute__((ext_vector_type(2))) float v2f;
typedef __attribute__((ext_vector_type(8))) float v8f;

#define EPS_BN 1e-5f
#define SLOPE 0.2f

// NTU hierarchy joint lists (LAYERS[i] = _G[i] + _G[i+1]), flattened.
__constant__ int LJ[45] = {
    1, 0, 20,                                  // L0 (3)
    0, 20, 12, 16, 2, 4, 8,                    // L1 (7)
    12, 16, 2, 4, 8, 13, 17, 3, 5, 9,          // L2 (10)
    13, 17, 3, 5, 9, 14, 18, 6, 10,            // L3 (9)
    14, 18, 6, 10, 15, 19, 7, 11,              // L4 (8)
    15, 19, 7, 11, 21, 22, 23, 24              // L5 (8)
};
__constant__ int LOFF[7] = {0, 3, 10, 20, 29, 37, 45};

// ---------------------------------------------------------------------------
// Kernel 1: x_t[n,c,l,v] = max_t x[n,c,l,t,v].   x: [N,C,L,T,V]=[32,256,6,64,25]
// One thread per (n,c,l,v); 25 adjacent lanes consume each contiguous
// 100-byte [v] row, so every cacheline fetched is (mostly) used.
// ---------------------------------------------------------------------------
__global__ void reduce_max_t(const float* __restrict__ x,
                             float* __restrict__ xt, int total) {
    int i = blockIdx.x * blockDim.x + threadIdx.x;   // over N*C*L*V
    if (i >= total) return;
    int v   = i % 25;
    int ncl = i / 25;
    const float* p = x + (size_t)ncl * (64 * 25) + v;
    float m = -__builtin_inff();
#pragma unroll 8
    for (int t = 0; t < 64; ++t) m = fmaxf(m, p[t * 25]);
    // xt flat index == (n*256+c)*150 + (l*25+v)  (spatial = l*25+v in [0,150))
    xt[i] = m;
}

// ---------------------------------------------------------------------------
// Kernel 2: h = relu(BN(W_down @ x_t + b_down)), via V_WMMA_F32_16X16X4_F32.
// Per sample n: D[64 x 150] = A[64 x 256] * B[256 x 150].
// grid = (n=32, Mtile=4, Ntile=10), block = 32 (one full wave; EXEC all 1s).
// ISA layouts (wave32):
//   A 16x4:  lane<16 -> row M=lane, K={k0,k0+1}; lane>=16 -> row M=lane-16,
//            K={k0+2,k0+3}  (VGPR0=Keven, VGPR1=Kodd)
//   B 4x16:  same K striping, N = lane%16
//   C/D 16x16: VGPR r, lane L -> M = r + 8*(L>=16), N = L%16
// BN is folded into the accumulator store (per-M affine) + ReLU.
// Out-of-range spatial columns: clamp the *address* only (valid memory);
// the polluted D columns belong to the same out-of-range lanes and are
// dropped by the store guard, so the K-loop stays divergence-free.
// ---------------------------------------------------------------------------
__global__ __launch_bounds__(32) void conv_down_wmma(
    const float* __restrict__ xt,    // [32][256][150]
    const float* __restrict__ W,     // [64][256]
    const float* __restrict__ bdn, const float* __restrict__ gdn,
    const float* __restrict__ bedn, const float* __restrict__ mdn,
    const float* __restrict__ vdn,
    float* __restrict__ h)           // [32][64][150]
{
    const int SP = 150;
    int n    = blockIdx.x;
    int mt   = blockIdx.y;           // 0..3
    int nt   = blockIdx.z;           // 0..9
    int lane = threadIdx.x;          // 0..31
    int hf   = lane >> 4;            // 0/1
    int l16  = lane & 15;
    int Mbase = mt * 16, Nbase = nt * 16;
    int col  = Nbase + l16;          // spatial column for B and D
    int ccol = col < SP ? col : (SP - 1);   // clamped (address-safe) column
    int arow = Mbase + l16;          // output-channel row for A
    int koff = hf * 2;

    const float* Bp = xt + (size_t)n * 256 * SP;

    v8f acc = {};
    for (int k0 = 0; k0 < 256; k0 += 4) {
        v2f a;
        a.x = W[arow * 256 + k0 + koff];
        a.y = W[arow * 256 + k0 + koff + 1];
        v2f b;
        b.x = Bp[(k0 + koff) * SP + ccol];
        b.y = Bp[(k0 + koff + 1) * SP + ccol];
        acc = __builtin_amdgcn_wmma_f32_16x16x4_f32(
            /*neg_a=*/false, a, /*neg_b=*/false, b,
            /*c_mod=*/(short)0, acc, /*reuse_a=*/false, /*reuse_b=*/false);
    }

    if (col < SP) {
#pragma unroll
        for (int r = 0; r < 8; ++r) {
            int m = Mbase + r + hf * 8;
            float scale = gdn[m] / sqrtf(vdn[m] + EPS_BN);
            float val = (acc[r] + bdn[m] - mdn[m]) * scale + bedn[m];
            h[((size_t)n * 64 + m) * SP + col] = fmaxf(val, 0.0f);
        }
    }
}

// ---------------------------------------------------------------------------
// Kernel 3: fused tiny graph ops per sample (one block / n, 256 threads):
//   s[c4][l]  = mean over joints of h        (384 items)
//   dist[i][j] (6x6) + stable top-3          (self included; dist[i][i]=0)
//   e[o][l]   = max_k leaky(BN(W_edge@gf))   (384 items, 128-length dots)
//   gate[c][l]= sigmoid(W_agg@e + b_agg)     (1536 items, 64-length dots)
// ---------------------------------------------------------------------------
__global__ __launch_bounds__(256) void graph_fused(
    const float* __restrict__ h,      // [32][64][150]
    const float* __restrict__ We,     // [64][128]
    const float* __restrict__ ge, const float* __restrict__ bee,
    const float* __restrict__ me, const float* __restrict__ ve,
    const float* __restrict__ Wagg,   // [256][64]
    const float* __restrict__ bagg,   // [256]
    float* __restrict__ gate)         // [32][256][6]
{
    __shared__ float s[64][6];
    __shared__ float dist[6][6];
    __shared__ int   knn[6][3];
    __shared__ float e[64][6];

    int n   = blockIdx.x;
    int tid = threadIdx.x;
    const float* hp = h + (size_t)n * 64 * 150;

    // ---- s: per-layer joint means ----
    for (int idx = tid; idx < 64 * 6; idx += 256) {
        int c4 = idx / 6, l = idx % 6;
        int o0 = LOFF[l], o1 = LOFF[l + 1];
        float sum = 0.0f;
        for (int j = o0; j < o1; ++j) sum += hp[c4 * 150 + l * 25 + LJ[j]];
        s[c4][l] = sum / (float)(o1 - o0);
    }
    __syncthreads();

    // ---- 6x6 inner products -> negative squared distances ----
    int di = tid / 6, dj = tid % 6;
    float inn = 0.0f;
    if (tid < 36) {
        for (int c = 0; c < 64; ++c) inn += s[c][di] * s[c][dj];
        dist[di][dj] = inn;
    }
    __syncthreads();
    float dval = 0.0f;
    if (tid < 36) dval = 2.0f * dist[di][dj] - dist[di][di] - dist[dj][dj];
    __syncthreads();
    if (tid < 36) dist[di][dj] = dval;
    __syncthreads();

    // ---- stable top-3 per row (ties -> lowest index, like lax.top_k) ----
    if (tid < 6) {
        bool used[6] = {false, false, false, false, false, false};
        for (int k = 0; k < 3; ++k) {
            float best = -__builtin_inff();
            int bi = 0;
            for (int j = 0; j < 6; ++j)
                if (!used[j] && dist[tid][j] > best) { best = dist[tid][j]; bi = j; }
            used[bi] = true;
            knn[tid][k] = bi;
        }
    }
    __syncthreads();

    // ---- EdgeConv + BN + leakyReLU + max over k ----
    for (int idx = tid; idx < 64 * 6; idx += 256) {
        int o = idx / 6, l = idx % 6;
        float scale = ge[o] / sqrtf(ve[o] + EPS_BN);
        float best = -__builtin_inff();
        for (int k = 0; k < 3; ++k) {
            int j = knn[l][k];
            float acc = 0.0f;
            for (int c = 0; c < 64; ++c) {
                float xe = s[c][l];
                acc += We[o * 128 + c] * (s[c][j] - xe)
                     + We[o * 128 + 64 + c] * xe;
            }
            float bn = (acc - me[o]) * scale + bee[o];
            float lr = bn > 0.0f ? bn : SLOPE * bn;
            best = fmaxf(best, lr);
        }
        e[o][l] = best;
    }
    __syncthreads();

    // ---- attention gate: sigmoid(W_agg @ e + b_agg) ----
    for (int idx = tid; idx < 256 * 6; idx += 256) {
        int c = idx / 6, l = idx % 6;
        float acc = bagg[c];
        for (int o = 0; o < 64; ++o) acc += Wagg[c * 64 + o] * e[o][l];
        gate[((size_t)n * 256 + c) * 6 + l] = 1.0f / (1.0f + __expf(-acc));
    }
}

// ---------------------------------------------------------------------------
// Kernel 4: out[n,c,t,v] = sum_l x[n,c,l,t,v] * gate[n,c,l]
// One thread per output element; 6 strided reads of x, gate hits cache.
// ---------------------------------------------------------------------------
__global__ void gate_sum(const float* __restrict__ x,
                         const float* __restrict__ gate,
                         float* __restrict__ out, int total) {
    int i = blockIdx.x * blockDim.x + threadIdx.x;   // over N*C*T*V
    if (i >= total) return;
    int tv = i % 1600;          // t*25+v
    int nc = i / 1600;          // n*256+c
    const float* xp = x + (size_t)nc * 9600 + tv;
    const float* gp = gate + (size_t)nc * 6;
    float acc = 0.0f;
#pragma unroll
    for (int l = 0; l < 6; ++l) acc += xp[l * 1600] * gp[l];
    out[i] = acc;
}

// ---------------------------------------------------------------------------
extern "C" void kernel_launch(void* const* d_in, const int* in_sizes, int n_in,
                              void* d_out, int out_size, void* d_ws, size_t ws_size,
                              hipStream_t stream) {
    const float* x      = (const float*)d_in[0];   // [32,256,6,64,25]
    const float* W_down = (const float*)d_in[1];   // [64,256]
    const float* b_down = (const float*)d_in[2];
    const float* g_down = (const float*)d_in[3];
    const float* be_down= (const float*)d_in[4];
    const float* m_down = (const float*)d_in[5];
    const float* v_down = (const float*)d_in[6];
    const float* W_edge = (const float*)d_in[7];   // [64,128]
    const float* g_edge = (const float*)d_in[8];
    const float* be_edge= (const float*)d_in[9];
    const float* m_edge = (const float*)d_in[10];
    const float* v_edge = (const float*)d_in[11];
    const float* W_agg  = (const float*)d_in[12];  // [256,64]
    const float* b_agg  = (const float*)d_in[13];  // [256]
    float* out = (float*)d_out;                    // [32,256,64,25]

    // workspace partition (floats): xt | h | gate
    float* xt_ws   = (float*)d_ws;                       // 32*256*150 = 1,228,800
    float* h_ws    = xt_ws + 32 * 256 * 150;             // 32*64*150  =   307,200
    float* gate_ws = h_ws + 32 * 64 * 150;               // 32*256*6   =    49,152

    // 1) max over T
    {
        int total = 32 * 256 * 6 * 25;
        reduce_max_t<<<(total + 255) / 256, 256, 0, stream>>>(x, xt_ws, total);
    }
    // 2) conv_down via WMMA (per-sample GEMM, one wave per 16x16 tile)
    {
        dim3 grid(32, 4, 10);
        conv_down_wmma<<<grid, 32, 0, stream>>>(xt_ws, W_down, b_down, g_down,
                                                be_down, m_down, v_down, h_ws);
    }
    // 3) fused hierarchy sampling + kNN EdgeConv + attention gate
    {
        graph_fused<<<32, 256, 0, stream>>>(h_ws, W_edge, g_edge, be_edge,
                                            m_edge, v_edge, W_agg, b_agg,
                                            gate_ws);
    }
    // 4) gated sum over hierarchy levels
    {
        int total = 32 * 256 * 64 * 25;
        gate_sum<<<(total + 255) / 256, 256, 0, stream>>>(x, gate_ws, out, total);
    }
}